// DipoleEnergyLean_66073776881930
// MI455X (gfx1250) — compile-verified
//
#include <hip/hip_runtime.h>
#include <math.h>

// ---------------- problem constants (match reference) ----------------
#define NA_   96              // atoms per molecule
#define ND_   384             // input features
#define NH_   192             // hidden
#define NF_   96              // species-MLP out
#define NS_   4               // species
#define CUT_  5.2f

#define MTILE 32              // atoms per block (2 x 16-row WMMA tiles)
#define LDA_  (ND_ + 4)       // 388 floats: row->bank conflict-free (388%64=4)
#define LDH_  (NH_ + 4)       // 196
#define LDI_  (NF_ + 4)       // 100

typedef __attribute__((ext_vector_type(2))) float v2f;
typedef __attribute__((ext_vector_type(8))) float v8f;

__device__ __forceinline__ v8f wmma_f32(v2f a, v2f b, v8f c) {
  // (neg_a, A, neg_b, B, c_mod, C, reuse_a, reuse_b)
  return __builtin_amdgcn_wmma_f32_16x16x4_f32(false, a, false, b, (short)0, c,
                                               false, false);
}

__device__ __forceinline__ float gelu_tanh(float x) {
  // jax.nn.gelu (approximate=True): 0.5x(1+tanh(u)) = x*sigmoid(2u), branch-free
  float u = 0.7978845608028654f * (x + 0.044715f * x * x * x);
  return x * __builtin_amdgcn_rcpf(1.0f + __expf(-2.0f * u));
}

// ---------------- kernel 1: init workspace ----------------
__global__ void init_kernel(int* cnt8, int* perm, float* merged,
                            int npad, long nmerged) {
  long idx = (long)blockIdx.x * blockDim.x + threadIdx.x;
  if (idx < 8) cnt8[idx] = 0;          // cnt[4] + cursor[4]
  if (idx < npad) perm[idx] = -1;
  if (idx < nmerged) merged[idx] = 0.0f;
}

// ---------------- kernel 2: species histogram ----------------
__global__ void hist_kernel(const int* __restrict__ species, int* cnt, int n) {
  int i = blockIdx.x * blockDim.x + threadIdx.x;
  if (i < n) atomicAdd(&cnt[species[i]], 1);
}

// ---------------- kernel 3: padded prefix (single thread, S=4) ----------------
__global__ void offsets_kernel(const int* __restrict__ cnt, int* offP) {
  int acc = 0;
  for (int s = 0; s < NS_; ++s) {
    offP[s] = acc;
    acc += ((cnt[s] + MTILE - 1) / MTILE) * MTILE;  // pad segments to 32 rows
  }
  offP[NS_] = acc;
}

// ---------------- kernel 4: scatter permutation ----------------
__global__ void scatter_kernel(const int* __restrict__ species,
                               const int* __restrict__ offP,
                               int* cursor, int* perm, int n) {
  int i = blockIdx.x * blockDim.x + threadIdx.x;
  if (i >= n) return;
  int s = species[i];
  int pos = offP[s] + atomicAdd(&cursor[s], 1);
  perm[pos] = i;
}

// ---------------- kernel 5: fused per-species MLP via WMMA ----------------
// One 32-atom tile per block (2 M-tiles -> B fragments reused across 2 WMMAs).
// 192 threads = 6 wave32s.
// GEMM1: [32,384]x[384,192] + GELU -> LDS
// GEMM2: [32,192]x[192, 96]        -> internal (global + LDS)
// GEMM3: [32, 96]x[ 96, 96]        -> nbr (global)
__global__ __launch_bounds__(192) void mlp_kernel(
    const float* __restrict__ x,        // [N, D]
    const float* __restrict__ W1,       // [S, D, H]
    const float* __restrict__ W2,       // [S, H, F]
    const float* __restrict__ Wn,       // [S, F, F]
    const int*   __restrict__ perm,     // [npad]
    const int*   __restrict__ offP,     // [S+1]
    float* __restrict__ internal_g,     // [N, F]
    float* __restrict__ nbr_g)          // [N, F]
{
  __shared__ float ldsA[MTILE * LDA_];  // ~48.5 KB  input rows
  __shared__ float ldsH[MTILE * LDH_];  // ~24.5 KB  gelu(x@W1)
  __shared__ float ldsI[MTILE * LDI_];  // ~12.5 KB  internal
  __shared__ int   ldsOrig[MTILE];
  __shared__ int   ldsSpec;

  const int t = blockIdx.x;
  const int totalTiles = offP[NS_] / MTILE;
  if (t >= totalTiles) return;          // uniform across block

  const int tid  = threadIdx.x;
  const int lane = tid & 31;
  const int wave = tid >> 5;            // 0..5

  if (tid < MTILE) ldsOrig[tid] = perm[t * MTILE + tid];
  if (tid == 0) {
    int s = 0;
#pragma unroll
    for (int i = 1; i < NS_; ++i)
      if (t * MTILE >= offP[i]) s = i;
    ldsSpec = s;
  }
  __syncthreads();
  const int s = ldsSpec;

  // ---- stage A tile into LDS (float4), zero for padded rows ----
  for (int i = tid; i < MTILE * (ND_ / 4); i += 192) {
    int row = i / (ND_ / 4);
    int c4  = i % (ND_ / 4);
    int orig = ldsOrig[row];
    float4 v = make_float4(0.f, 0.f, 0.f, 0.f);
    if (orig >= 0)
      v = reinterpret_cast<const float4*>(x + (size_t)orig * ND_)[c4];
    reinterpret_cast<float4*>(&ldsA[row * LDA_])[c4] = v;
  }
  __syncthreads();

  // WMMA f32 16x16x4 lane mapping (ISA 7.12.2):
  //  A: lane L<16 -> row L, VGPR0 K=k0+0, VGPR1 K=k0+1;  lane>=16 -> K offset +2
  //  B: VGPR0 lanes0-15 K=k0+0 N=lane; lanes16-31 K=k0+2; VGPR1 rows +1
  //  C: VGPR v: lanes0-15 M=v, lanes16-31 M=v+8, N=lane&15
  const int m     = lane & 15;
  const int koff  = (lane >> 4) << 1;   // 0 or 2
  const int col   = lane & 15;
  const int rbase = (lane >> 4) << 3;   // 0 or 8

  // ---- GEMM1 + GELU : 12 N-tiles over H=192, 6 waves x 2 n-tiles ----
  const float* W1s = W1 + (size_t)s * ND_ * NH_;
  for (int ntile = wave; ntile < NH_ / 16; ntile += 6) {
    const int n0 = ntile * 16;
    v8f acc0 = {}, acc1 = {};
#pragma unroll 4
    for (int k0 = 0; k0 < ND_; k0 += 4) {
      const int kk = k0 + koff;
      v2f b, a0, a1;
      b.x  = W1s[(size_t)kk * NH_ + n0 + col];
      b.y  = W1s[(size_t)(kk + 1) * NH_ + n0 + col];
      a0.x = ldsA[m * LDA_ + kk];
      a0.y = ldsA[m * LDA_ + kk + 1];
      a1.x = ldsA[(m + 16) * LDA_ + kk];
      a1.y = ldsA[(m + 16) * LDA_ + kk + 1];
      acc0 = wmma_f32(a0, b, acc0);
      acc1 = wmma_f32(a1, b, acc1);
    }
#pragma unroll
    for (int v = 0; v < 8; ++v) {
      ldsH[(rbase + v) * LDH_ + n0 + col]        = gelu_tanh(acc0[v]);
      ldsH[(rbase + v + 16) * LDH_ + n0 + col]   = gelu_tanh(acc1[v]);
    }
  }
  __syncthreads();

  // ---- GEMM2 : 6 N-tiles over F=96, one per wave ----
  const float* W2s = W2 + (size_t)s * NH_ * NF_;
  {
    const int n0 = wave * 16;
    v8f acc0 = {}, acc1 = {};
#pragma unroll 4
    for (int k0 = 0; k0 < NH_; k0 += 4) {
      const int kk = k0 + koff;
      v2f b, a0, a1;
      b.x  = W2s[(size_t)kk * NF_ + n0 + col];
      b.y  = W2s[(size_t)(kk + 1) * NF_ + n0 + col];
      a0.x = ldsH[m * LDH_ + kk];
      a0.y = ldsH[m * LDH_ + kk + 1];
      a1.x = ldsH[(m + 16) * LDH_ + kk];
      a1.y = ldsH[(m + 16) * LDH_ + kk + 1];
      acc0 = wmma_f32(a0, b, acc0);
      acc1 = wmma_f32(a1, b, acc1);
    }
#pragma unroll
    for (int v = 0; v < 8; ++v) {
      const int r0 = rbase + v, r1 = rbase + v + 16;
      ldsI[r0 * LDI_ + n0 + col] = acc0[v];
      ldsI[r1 * LDI_ + n0 + col] = acc1[v];
      const int o0 = ldsOrig[r0], o1 = ldsOrig[r1];
      if (o0 >= 0) internal_g[(size_t)o0 * NF_ + n0 + col] = acc0[v];
      if (o1 >= 0) internal_g[(size_t)o1 * NF_ + n0 + col] = acc1[v];
    }
  }
  __syncthreads();

  // ---- GEMM3 (Wn) : 6 N-tiles over F=96, one per wave ----
  const float* Wns = Wn + (size_t)s * NF_ * NF_;
  {
    const int n0 = wave * 16;
    v8f acc0 = {}, acc1 = {};
#pragma unroll 4
    for (int k0 = 0; k0 < NF_; k0 += 4) {
      const int kk = k0 + koff;
      v2f b, a0, a1;
      b.x  = Wns[(size_t)kk * NF_ + n0 + col];
      b.y  = Wns[(size_t)(kk + 1) * NF_ + n0 + col];
      a0.x = ldsI[m * LDI_ + kk];
      a0.y = ldsI[m * LDI_ + kk + 1];
      a1.x = ldsI[(m + 16) * LDI_ + kk];
      a1.y = ldsI[(m + 16) * LDI_ + kk + 1];
      acc0 = wmma_f32(a0, b, acc0);
      acc1 = wmma_f32(a1, b, acc1);
    }
#pragma unroll
    for (int v = 0; v < 8; ++v) {
      const int o0 = ldsOrig[rbase + v], o1 = ldsOrig[rbase + v + 16];
      if (o0 >= 0) nbr_g[(size_t)o0 * NF_ + n0 + col] = acc0[v];
      if (o1 >= 0) nbr_g[(size_t)o1 * NF_ + n0 + col] = acc1[v];
    }
  }
}

// ---------------- kernel 6: edge decay weights ----------------
__global__ void decay_kernel(const float* __restrict__ dist,
                             const float* __restrict__ pref,
                             const float* __restrict__ fac,
                             float* __restrict__ decay, int E) {
  int e = blockIdx.x * blockDim.x + threadIdx.x;
  if (e >= E) return;
  float d = dist[e];
  float xx = (CUT_ - d) * (1.0f / CUT_);
  xx = fminf(fmaxf(xx, 0.0f), 1.0f);
  float sc = xx * xx * xx * (xx * (6.0f * xx - 15.0f) + 10.0f);
  float p = pref[0], f = fac[0];
  decay[e] = p * p * __expf(-(f * f) * d) * sc;
}

// ---------------- kernel 7: edge gather + scatter-add ----------------
// 48 threads per edge: 2 directions x 24 float4 chunks of F=96.
__global__ __launch_bounds__(256) void edge_kernel(
    const int*   __restrict__ ai,      // [2, E]
    const float* __restrict__ decay,   // [E]
    const float* __restrict__ nbr,     // [N, F]
    float*       __restrict__ merged,  // [N, F]
    int E) {
  size_t idx = (size_t)blockIdx.x * blockDim.x + threadIdx.x;
  size_t total = (size_t)E * 48;
  if (idx >= total) return;
  int e = (int)(idx / 48);
  int r = (int)(idx % 48);
  int i0 = ai[e];
  int i1 = ai[E + e];
  int src = (r < 24) ? i1 : i0;
  int dst = (r < 24) ? i0 : i1;
  int f0  = (r % 24) * 4;
  float w = decay[e];
  float4 v = *reinterpret_cast<const float4*>(nbr + (size_t)src * NF_ + f0);
  float* mp = merged + (size_t)dst * NF_ + f0;
  unsafeAtomicAdd(mp + 0, v.x * w);
  unsafeAtomicAdd(mp + 1, v.y * w);
  unsafeAtomicAdd(mp + 2, v.z * w);
  unsafeAtomicAdd(mp + 3, v.w * w);
}

// ---------------- kernel 8: head + per-molecule redistribution ----------------
__global__ __launch_bounds__(128) void head_kernel(
    const int*   __restrict__ species,
    const float* __restrict__ internal_g,   // [N, F]
    const float* __restrict__ merged,       // [N, F]
    const float* __restrict__ Wf,           // [S, 2F, 1]
    const float* __restrict__ total_charges,// [B]
    float* __restrict__ out, int N) {
  __shared__ float sSum[128];
  __shared__ float sCnt[128];
  const int b = blockIdx.x;
  const int a = threadIdx.x;
  const int i = b * NA_ + a;

  float pre = 0.0f;
  float cntv = 0.0f;
  int spv = 0;
  if (a < NA_) {
    spv = species[i];
    cntv = (spv != -1) ? 1.0f : 0.0f;
    const float* wf = Wf + (size_t)spv * 2 * NF_;
    const float* iv = internal_g + (size_t)i * NF_;
    const float* mv = merged + (size_t)i * NF_;
    float acc = 0.0f;
#pragma unroll 4
    for (int k = 0; k < NF_; ++k) acc = fmaf(iv[k], wf[k], acc);
#pragma unroll 4
    for (int k = 0; k < NF_; ++k) acc = fmaf(mv[k], wf[NF_ + k], acc);
    pre = acc;
  }
  sSum[a] = (a < NA_) ? pre : 0.0f;
  sCnt[a] = cntv;
  __syncthreads();
  for (int off = 64; off > 0; off >>= 1) {
    if (a < off) { sSum[a] += sSum[a + off]; sCnt[a] += sCnt[a + off]; }
    __syncthreads();
  }
  const float ssum = sSum[0];
  const float cnt  = sCnt[0];

  if (a < NA_) {
    float factors = 1.0f / cnt;
    float ch = pre + (total_charges[b] - ssum) * factors;
    if (spv == -1) ch = 0.0f;
    reinterpret_cast<int*>(out)[i] = spv;  // output 0: species (raw int bits)
    out[N + i]     = ch;                   // output 1: charges
    out[2 * N + i] = pre;                  // output 2: precharges
  }
}

// ---------------- host ----------------
extern "C" void kernel_launch(void* const* d_in, const int* in_sizes, int n_in,
                              void* d_out, int out_size, void* d_ws, size_t ws_size,
                              hipStream_t stream) {
  const int*   species = (const int*)  d_in[0];
  const float* x       = (const float*)d_in[1];
  const int*   ai      = (const int*)  d_in[2];
  const float* dist    = (const float*)d_in[3];
  const float* totq    = (const float*)d_in[4];
  const float* W1      = (const float*)d_in[5];
  const float* W2      = (const float*)d_in[6];
  const float* Wn      = (const float*)d_in[7];
  const float* Wf      = (const float*)d_in[8];
  const float* pref    = (const float*)d_in[9];
  const float* fac     = (const float*)d_in[10];

  const int N = in_sizes[0];          // B*A = 24576
  const int E = in_sizes[3];          // edges
  const int Bm = N / NA_;             // molecules
  const int npad = N + NS_ * MTILE;   // worst-case padded rows
  const int maxTiles = npad / MTILE;

  // workspace carve-up (256B aligned)
  char* ws = (char*)d_ws;
  int*   cnt      = (int*)(ws + 0);          // [4]
  int*   cursor   = (int*)(ws + 16);         // [4]
  int*   offP     = (int*)(ws + 64);         // [5]
  int*   perm     = (int*)(ws + 256);        // [npad]
  size_t off = 256 + ((size_t)npad * 4 + 255) / 256 * 256;
  float* decay    = (float*)(ws + off);  off += ((size_t)E * 4 + 255) / 256 * 256;
  float* internal = (float*)(ws + off);  off += (size_t)N * NF_ * 4;
  float* nbr      = (float*)(ws + off);  off += (size_t)N * NF_ * 4;
  float* merged   = (float*)(ws + off);  off += (size_t)N * NF_ * 4;
  (void)ws_size; (void)n_in; (void)out_size;

  const long nmerged = (long)N * NF_;
  init_kernel<<<(unsigned)((nmerged + 255) / 256), 256, 0, stream>>>(
      cnt, perm, merged, npad, nmerged);
  hist_kernel<<<(N + 255) / 256, 256, 0, stream>>>(species, cnt, N);
  offsets_kernel<<<1, 1, 0, stream>>>(cnt, offP);
  scatter_kernel<<<(N + 255) / 256, 256, 0, stream>>>(species, offP, cursor, perm, N);
  mlp_kernel<<<maxTiles, 192, 0, stream>>>(x, W1, W2, Wn, perm, offP,
                                           internal, nbr);
  decay_kernel<<<(E + 255) / 256, 256, 0, stream>>>(dist, pref, fac, decay, E);
  {
    size_t tot = (size_t)E * 48;
    edge_kernel<<<(unsigned)((tot + 255) / 256), 256, 0, stream>>>(
        ai, decay, nbr, merged, E);
  }
  head_kernel<<<Bm, 128, 0, stream>>>(species, internal, merged, Wf, totq,
                                      (float*)d_out, N);
}